// ListGen_ANN_47382079209946
// MI455X (gfx1250) — compile-verified
//
#include <hip/hip_runtime.h>
#include <stdint.h>

#define C_DIM 64
#define N_DIM 250
#define D_DIM 512
#define K_DIM 8
#define SIGMA 0.05f

// ---------------------------------------------------------------------------
// Helpers: CDNA5 async global->LDS copy of one 2KB row (512 f32), per wave.
// Each of the 32 lanes copies 16B at 4 instruction offsets; INST_OFFSET is
// added to BOTH the LDS address and the global address, so the LDS buffer
// ends up an exact contiguous copy of the row. Tracked by ASYNCcnt.
// ---------------------------------------------------------------------------
__device__ __forceinline__ void async_copy_row_to_lds(const float* gsrc,
                                                      unsigned lds_byte_off,
                                                      int lane) {
    unsigned ldsa = lds_byte_off + (unsigned)(lane * 16);
    unsigned voff = (unsigned)(lane * 16);
    asm volatile(
        "global_load_async_to_lds_b128 %0, %1, %2 offset:0\n"
        "global_load_async_to_lds_b128 %0, %1, %2 offset:512\n"
        "global_load_async_to_lds_b128 %0, %1, %2 offset:1024\n"
        "global_load_async_to_lds_b128 %0, %1, %2 offset:1536\n"
        :: "v"(ldsa), "v"(voff), "s"(gsrc)
        : "memory");
}

template <int N>
__device__ __forceinline__ void wait_asynccnt() {
    asm volatile("s_wait_asynccnt %0" :: "i"(N) : "memory");
}

// Low 32 bits of a generic pointer to a __shared__ object == LDS byte offset.
__device__ __forceinline__ unsigned lds_offset_of(const void* p) {
    return (unsigned)(uintptr_t)p;
}

// ---------------------------------------------------------------------------
// Kernel 1: per-row bitonic sort (descending value, ascending index ties).
// Emits x_sorted[c, p] and inv[c, orig] = p   (matches stable argsort(-x)).
// ---------------------------------------------------------------------------
__global__ __launch_bounds__(D_DIM) void sort_rows(const float* __restrict__ x,
                                                   float* __restrict__ x_sorted,
                                                   int* __restrict__ inv) {
    __shared__ float sv[D_DIM];
    __shared__ int   si[D_DIM];
    const int c = blockIdx.x;
    const int t = threadIdx.x;

    sv[t] = x[c * D_DIM + t];
    si[t] = t;
    __syncthreads();

    for (unsigned k = 2; k <= D_DIM; k <<= 1) {
        for (unsigned j = k >> 1; j > 0; j >>= 1) {
            unsigned ixj = t ^ j;
            if (ixj > (unsigned)t) {
                float a = sv[t], b = sv[ixj];
                int   ai = si[t], bi = si[ixj];
                // total order for descending sort: a before b
                bool aFirst = (a > b) || (a == b && ai < bi);
                bool desc = ((t & k) == 0u);
                if (desc ? !aFirst : aFirst) {
                    sv[t] = b; sv[ixj] = a;
                    si[t] = bi; si[ixj] = ai;
                }
            }
            __syncthreads();
        }
    }

    x_sorted[c * D_DIM + t] = sv[t];
    inv[c * D_DIM + si[t]] = t;        // inverse permutation
}

// ---------------------------------------------------------------------------
// Kernel 2: per-sample top-8 + histogram.
// Grid: C*4 blocks (4 blocks per c), 256 threads = 8 wave32.
// Each wave processes samples n = gw, gw+32, ... with double-buffered
// async global->LDS streaming of the 2KB noise row.
// ---------------------------------------------------------------------------
__global__ __launch_bounds__(256) void topk_hist(const float* __restrict__ noise,
                                                 const float* __restrict__ x_sorted,
                                                 unsigned* __restrict__ gcounts) {
    __shared__ float    xs[D_DIM];             //  2 KB
    __shared__ unsigned scnt[D_DIM * K_DIM];   // 16 KB
    __shared__ float    nbuf[8][2][D_DIM];     // 32 KB (per-wave double buffer)

    const int c    = blockIdx.x >> 2;
    const int s    = blockIdx.x & 3;
    const int tid  = threadIdx.x;
    const int w    = tid >> 5;
    const int lane = tid & 31;
    const int gw   = s * 8 + w;                // 0..31 over the 4-block group

    // stage x_sorted row, zero histogram
    xs[tid]       = x_sorted[c * D_DIM + tid];
    xs[tid + 256] = x_sorted[c * D_DIM + tid + 256];
    #pragma unroll
    for (int i = 0; i < 16; ++i) scnt[tid + 256 * i] = 0u;
    __syncthreads();

    const float* nrow_base = noise + (size_t)c * N_DIM * D_DIM;
    const unsigned buf_off[2] = { lds_offset_of(&nbuf[w][0][0]),
                                  lds_offset_of(&nbuf[w][1][0]) };

    // prologue: kick off first sample for this wave (gw < 32 <= 250 always)
    async_copy_row_to_lds(nrow_base + (size_t)gw * D_DIM, buf_off[0], lane);

    int cur = 0;
    for (int n = gw; n < N_DIM; n += 32) {
        int nn = n + 32;
        if (nn < N_DIM) {
            async_copy_row_to_lds(nrow_base + (size_t)nn * D_DIM,
                                  buf_off[cur ^ 1], lane);
            wait_asynccnt<4>();   // 8 outstanding -> first 4 (cur buf) done
        } else {
            wait_asynccnt<0>();
        }

        const float* nb = &nbuf[w][cur][0];

        // lane holds elements e = lane + 32*i (conflict-free LDS reads)
        float vals[16];
        #pragma unroll
        for (int i = 0; i < 16; ++i)
            vals[i] = nb[lane + 32 * i] * SIGMA + xs[lane + 32 * i];

        // 8x argmax extraction; tie -> smaller index (matches lax.top_k)
        unsigned removed = 0u;
        int mysel = -1;                       // lane it keeps winner of iter it
        #pragma unroll
        for (int it = 0; it < K_DIM; ++it) {
            float bv = -3.4e38f;
            int   bi = 1 << 20;
            #pragma unroll
            for (int i = 0; i < 16; ++i) {
                int  id = lane + (i << 5);
                bool ok = ((removed >> i) & 1u) == 0u;
                bool better = ok && ((vals[i] > bv) || (vals[i] == bv && id < bi));
                bv = better ? vals[i] : bv;
                bi = better ? id : bi;
            }
            #pragma unroll
            for (int off = 16; off > 0; off >>= 1) {
                float ov = __shfl_xor(bv, off);
                int   oi = __shfl_xor(bi, off);
                bool take = (ov > bv) || (ov == bv && oi < bi);
                bv = take ? ov : bv;
                bi = take ? oi : bi;
            }
            if (lane == (bi & 31)) removed |= (1u << (bi >> 5));
            if (lane == it) mysel = bi;
        }

        // rank of each winner by ascending index (== jnp.sort(arg_locs))
        int rank = 0;
        #pragma unroll
        for (int j = 0; j < K_DIM; ++j) {
            int oj = __shfl(mysel, j);
            rank += (oj < mysel) ? 1 : 0;
        }
        if (lane < K_DIM)
            atomicAdd(&scnt[(mysel << 3) + rank], 1u);   // ds_add_u32

        cur ^= 1;
    }

    __syncthreads();
    // flush block-partial histogram to the global per-c histogram
    for (int e = tid; e < D_DIM * K_DIM; e += 256) {
        unsigned v = scnt[e];
        if (v) atomicAdd(&gcounts[(size_t)c * D_DIM * K_DIM + e], v);
    }
}

// ---------------------------------------------------------------------------
// Kernel 3: out[c,j,k] = counts[c, inv[c,j], k] / N
// ---------------------------------------------------------------------------
__global__ __launch_bounds__(256) void finalize(const unsigned* __restrict__ gcounts,
                                                const int* __restrict__ inv,
                                                float* __restrict__ out) {
    int t = blockIdx.x * 256 + threadIdx.x;          // < 64*512*8
    int c = t >> 12;
    int j = (t >> 3) & (D_DIM - 1);
    int k = t & 7;
    int p = inv[c * D_DIM + j];
    out[t] = (float)gcounts[(size_t)c * D_DIM * K_DIM + p * K_DIM + k]
             * (1.0f / (float)N_DIM);
}

// ---------------------------------------------------------------------------
extern "C" void kernel_launch(void* const* d_in, const int* in_sizes, int n_in,
                              void* d_out, int out_size, void* d_ws, size_t ws_size,
                              hipStream_t stream) {
    const float* x     = (const float*)d_in[0];   // [64,512]
    const float* noise = (const float*)d_in[1];   // [64,250,512]
    float* out = (float*)d_out;                   // [64,512,8]

    char* ws = (char*)d_ws;
    float*    x_sorted = (float*)(ws);                               // 128 KB
    int*      inv      = (int*)(ws + 131072);                        // 128 KB
    unsigned* gcounts  = (unsigned*)(ws + 262144);                   //   1 MB

    hipMemsetAsync(gcounts, 0, (size_t)C_DIM * D_DIM * K_DIM * sizeof(unsigned),
                   stream);

    sort_rows<<<C_DIM, D_DIM, 0, stream>>>(x, x_sorted, inv);
    topk_hist<<<C_DIM * 4, 256, 0, stream>>>(noise, x_sorted, gcounts);
    finalize<<<(C_DIM * D_DIM * K_DIM) / 256, 256, 0, stream>>>(gcounts, inv, out);
}